// Siren_816043786257
// MI455X (gfx1250) — compile-verified
//
#include <hip/hip_runtime.h>

typedef __attribute__((ext_vector_type(2))) float v2f;
typedef __attribute__((ext_vector_type(8))) float v8f;

#define SLOTS 252          // floats of constant image per lane
#define LANE_STRIDE 260    // LDS stride (floats): 260%64==4 -> 2-way-optimal b128, 16B aligned
#define OMEGA_OVER_2PI 4.774648292756860f   // 30 / (2*pi), v_sin takes revolutions
#define LOG2E 1.4426950408889634f

// ---------- small device helpers ----------

// Pure-VALU half-wave swap (lane <-> lane^16): v_permlanex16_b32 with identity
// nibble selectors. Avoids the DS pipe + s_wait_dscnt on the layer-to-layer
// critical path (vs ds_swizzle SWAPX16).
__device__ __forceinline__ float swz16(float x) {
  int xi = __float_as_int(x);
  int r = __builtin_amdgcn_permlanex16(xi, xi, 0x76543210, 0xfedcba98, false, false);
  return __int_as_float(r);
}

__device__ __forceinline__ v8f wmma4(v2f a, v2f b, v8f c) {
  // D(16x16,f32) = A(16x4,f32) * B(4x16,f32) + C ; full fp32 matrix pipe
  return __builtin_amdgcn_wmma_f32_16x16x4_f32(false, a, false, b, (short)0, c,
                                               false, false);
}

__device__ __forceinline__ v2f ld2(const float* p) { return *(const v2f*)p; }

__device__ __forceinline__ v8f ld8(const float* p) {
  float4 a = *(const float4*)(p);
  float4 b = *(const float4*)(p + 4);
  v8f r;
  r[0] = a.x; r[1] = a.y; r[2] = a.z; r[3] = a.w;
  r[4] = b.x; r[5] = b.y; r[6] = b.z; r[7] = b.w;
  return r;
}

__device__ __forceinline__ v8f act_sin(v8f x) {
  v8f r = x;
#pragma unroll
  for (int i = 0; i < 8; ++i)
    r[i] = __builtin_amdgcn_sinf(x[i] * OMEGA_OVER_2PI);  // sin(30*x)
  return r;
}

__device__ __forceinline__ float sigmoid_fast(float x) {
  float t = __builtin_amdgcn_exp2f(-x * LOG2E);
  return __builtin_amdgcn_rcpf(1.0f + t);
}

// Convert one 16-feature D-bank (feat = r + 8*(lane>=16), point = lane&15)
// into 4 B-chunks (K=4 each) of the next layer's B matrix.
// Permutes run unconditionally (full EXEC) BEFORE the divergent selects.
__device__ __forceinline__ void buildB(const v8f d, v2f* bc, bool lo) {
  float s0 = swz16(d[0]), s1 = swz16(d[1]), s2 = swz16(d[2]), s3 = swz16(d[3]);
  float s4 = swz16(d[4]), s5 = swz16(d[5]), s6 = swz16(d[6]), s7 = swz16(d[7]);
  bc[0].x = lo ? d[0] : s2;  bc[0].y = lo ? d[1] : s3;   // feats 0..3
  bc[1].x = lo ? d[4] : s6;  bc[1].y = lo ? d[5] : s7;   // feats 4..7
  bc[2].x = lo ? s0 : d[2];  bc[2].y = lo ? s1 : d[3];   // feats 8..11
  bc[3].x = lo ? s4 : d[6];  bc[3].y = lo ? s5 : d[7];   // feats 12..15
}

// 32-K accumulation: C from LDS (bias), 8 chained WMMAs with A chunks from LDS.
// (Compiler hoists these loop-invariant LDS loads into high VGPRs.)
__device__ __forceinline__ v8f gemm8(const float* cp, int aOff, int cOff,
                                     const v2f* bc) {
  v8f acc = ld8(cp + cOff);
#pragma unroll
  for (int j = 0; j < 8; ++j)
    acc = wmma4(ld2(cp + aOff + 2 * j), bc[j], acc);
  return acc;
}

// ---------- prep kernel: build per-lane WMMA-layout constant image ----------
// Layout per lane (floats):
//   [0,4)    L0 A (a_lo e0,e1 ; a_hi e0,e1)          (K=2 padded to 4)
//   [4,20)   L0 bias as C (lo 8, hi 8)
//   [20,212) L1..L4: {A_lo 16, A_hi 16, C_lo 8, C_hi 8} x 4
//   [212,236) L5: A_lo 16, C_lo 8
//   [236,252) final: A 8 (rows>=3 zero), C 8 (bias rows<3, lo lanes)
__global__ void siren_prep(float* __restrict__ cimg,
                           const float* __restrict__ w0, const float* __restrict__ b0,
                           const float* __restrict__ w1, const float* __restrict__ b1,
                           const float* __restrict__ w2, const float* __restrict__ b2,
                           const float* __restrict__ w3, const float* __restrict__ b3,
                           const float* __restrict__ w4, const float* __restrict__ b4,
                           const float* __restrict__ w5, const float* __restrict__ b5,
                           const float* __restrict__ wf, const float* __restrict__ bf) {
  int idx = blockIdx.x * blockDim.x + threadIdx.x;
  if (idx >= 32 * SLOTS) return;
  int lane = idx / SLOTS;
  int s = idx % SLOTS;
  int half = lane >> 4, l16 = lane & 15;
  float v = 0.0f;
  if (s < 4) {                                   // L0 A-chunks (in=2, K padded to 4)
    int mo = (s >> 1) * 16, e = s & 1, k = 2 * half + e;
    v = (k < 2) ? w0[(mo + l16) * 2 + k] : 0.0f;
  } else if ((s -= 4) < 16) {                    // L0 bias -> C layout
    int mo = (s >> 3) * 16, r = s & 7;
    v = b0[mo + r + 8 * half];
  } else if ((s -= 16) < 192) {                  // layers 1..4
    int L = s / 48, t = s % 48;
    const float* w = (L == 0) ? w1 : (L == 1) ? w2 : (L == 2) ? w3 : w4;
    const float* b = (L == 0) ? b1 : (L == 1) ? b2 : (L == 2) ? b3 : b4;
    if (t < 32) {                                // A chunks (lo then hi out-rows)
      int mo = (t >> 4) * 16, i = t & 15, j = i >> 1, e = i & 1;
      int k = 4 * j + 2 * half + e;
      v = w[(mo + l16) * 32 + k];
    } else {                                     // bias -> C layout
      int u = t - 32, mo = (u >> 3) * 16, r = u & 7;
      v = b[mo + r + 8 * half];
    }
  } else if ((s -= 192) < 24) {                  // layer 5 (32->16)
    if (s < 16) {
      int j = s >> 1, e = s & 1, k = 4 * j + 2 * half + e;
      v = w5[l16 * 32 + k];
    } else {
      int r = s - 16;
      v = b5[r + 8 * half];
    }
  } else {                                       // final linear (16->3)
    s -= 24;
    if (s < 8) {
      int j = s >> 1, e = s & 1, k = 4 * j + 2 * half + e;
      v = (l16 < 3) ? wf[l16 * 16 + k] : 0.0f;
    } else {
      int r = s - 8;
      v = (half == 0 && r < 3) ? bf[r] : 0.0f;
    }
  }
  cimg[idx] = v;
}

// ---------- main kernel: one wave processes 16 points end-to-end ----------
__global__ void __launch_bounds__(256) siren_fwd(const float2* __restrict__ coords,
                                                 const float* __restrict__ cimg,
                                                 float* __restrict__ out,
                                                 int ntiles, int npts) {
  __shared__ float lds[32 * LANE_STRIDE];
  for (int f = threadIdx.x; f < 32 * SLOTS; f += blockDim.x)
    lds[(f / SLOTS) * LANE_STRIDE + (f % SLOTS)] = cimg[f];
  __syncthreads();

  const int lane = threadIdx.x & 31;
  const bool lo = lane < 16;
  const float* cp = &lds[lane * LANE_STRIDE];

  const int wavesPerBlk = blockDim.x >> 5;
  const int gw = blockIdx.x * wavesPerBlk + (threadIdx.x >> 5);
  const int nw = gridDim.x * wavesPerBlk;

  for (int tile = gw; tile < ntiles; tile += nw) {
    int pid = tile * 16 + (lane & 15);
    int pc = pid < npts ? pid : npts - 1;
    float2 xy = coords[pc];

    // layer 0: B = coords (K=2 padded to 4, hi-lanes carry k=2,3 = 0)
    v2f bxy;
    bxy.x = lo ? xy.x : 0.0f;
    bxy.y = lo ? xy.y : 0.0f;
    v8f Dlo = act_sin(wmma4(ld2(cp + 0), bxy, ld8(cp + 4)));
    v8f Dhi = act_sin(wmma4(ld2(cp + 2), bxy, ld8(cp + 12)));

    // layers 1..4 (32 -> 32)
#pragma unroll
    for (int L = 0; L < 4; ++L) {
      const int base = 20 + 48 * L;
      v2f bc[8];
      buildB(Dlo, bc, lo);       // in-feats 0..15  -> chunks 0..3
      buildB(Dhi, bc + 4, lo);   // in-feats 16..31 -> chunks 4..7
      v8f nLo = gemm8(cp, base, base + 32, bc);
      v8f nHi = gemm8(cp, base + 16, base + 40, bc);
      Dlo = act_sin(nLo);
      Dhi = act_sin(nHi);
    }

    // layer 5 (32 -> 16)
    v2f bc5[8];
    buildB(Dlo, bc5, lo);
    buildB(Dhi, bc5 + 4, lo);
    v8f x5 = act_sin(gemm8(cp, 212, 228, bc5));

    // final linear (16 -> 3) + sigmoid
    v2f bcf[4];
    buildB(x5, bcf, lo);
    v8f accF = ld8(cp + 244);
#pragma unroll
    for (int j = 0; j < 4; ++j)
      accF = wmma4(ld2(cp + 236 + 2 * j), bcf[j], accF);

    float y0 = sigmoid_fast(accF[0]);
    float y1 = sigmoid_fast(accF[1]);
    float y2 = sigmoid_fast(accF[2]);

    if (lo && pid < npts) {
      float* o = out + 3 * (size_t)pid;
      o[0] = y0;
      o[1] = y1;
      o[2] = y2;
    }
  }
}

// ---------- launcher ----------
extern "C" void kernel_launch(void* const* d_in, const int* in_sizes, int n_in,
                              void* d_out, int out_size, void* d_ws, size_t ws_size,
                              hipStream_t stream) {
  (void)n_in; (void)out_size; (void)ws_size;
  const float* coords = (const float*)d_in[0];
  const float* w0 = (const float*)d_in[1];
  const float* b0 = (const float*)d_in[2];
  const float* w1 = (const float*)d_in[3];
  const float* b1 = (const float*)d_in[4];
  const float* w2 = (const float*)d_in[5];
  const float* b2 = (const float*)d_in[6];
  const float* w3 = (const float*)d_in[7];
  const float* b3 = (const float*)d_in[8];
  const float* w4 = (const float*)d_in[9];
  const float* b4 = (const float*)d_in[10];
  const float* w5 = (const float*)d_in[11];
  const float* b5 = (const float*)d_in[12];
  const float* wf = (const float*)d_in[13];
  const float* bf = (const float*)d_in[14];

  float* cimg = (float*)d_ws;   // 32*252*4 = 32256 bytes
  int npts = in_sizes[0] / 2;
  int ntiles = (npts + 15) / 16;

  siren_prep<<<(32 * SLOTS + 255) / 256, 256, 0, stream>>>(
      cimg, w0, b0, w1, b1, w2, b2, w3, b3, w4, b4, w5, b5, wf, bf);

  siren_fwd<<<2048, 256, 0, stream>>>((const float2*)coords, cimg,
                                      (float*)d_out, ntiles, npts);
}